// HyperConnection_54262616818195
// MI455X (gfx1250) — compile-verified
//
#include <hip/hip_runtime.h>
#include <math.h>

typedef __attribute__((ext_vector_type(2))) float v2f;
typedef __attribute__((ext_vector_type(8))) float v8f;

#define B_  4
#define S_  4096
#define NS  4
#define D_  1024
#define ND  (NS * D_)          // 4096 columns per (b,s)
#define NCHUNK 16              // S split into 16 chunks of 256
#define NDELTA 24              // n*n + 2n
#define NBS (B_ * S_)          // 16384 (b,s) rows

// ---------------------------------------------------------------------------
// Phase 1: partial sums of H over S.  partial[(b*16+chunk)*4096 + c]
// grid: (4 col-groups, 16 s-chunks, B), block: 256 threads, float4 per thread
// ---------------------------------------------------------------------------
__global__ __launch_bounds__(256)
void k_partial(const float* __restrict__ H, float* __restrict__ partial) {
    const int g     = blockIdx.x;          // column group (0..3), 1024 cols each
    const int chunk = blockIdx.y;          // s chunk (0..15), 256 s each
    const int b     = blockIdx.z;
    const int tid   = threadIdx.x;
    const int col   = g * 1024 + tid * 4;

    long base = ((long)b * S_ + (long)chunk * 256) * ND + col;
    float4 acc = make_float4(0.f, 0.f, 0.f, 0.f);
    #pragma unroll 4
    for (int s = 0; s < 256; ++s) {
        float4 v = *(const float4*)(H + base + (long)s * ND);
        acc.x += v.x; acc.y += v.y; acc.z += v.z; acc.w += v.w;
    }
    *(float4*)(partial + ((long)(b * NCHUNK + chunk)) * ND + col) = acc;
}

// ---------------------------------------------------------------------------
// Phase 2: finish mean, RMS-norm, delta = x@phi, gates + sinkhorn.
// grid: B blocks, 256 threads. Writes W[b][16][4] (padded) and hpost[b][4].
// ---------------------------------------------------------------------------
__global__ __launch_bounds__(256)
void k_gates(const float* __restrict__ partial, const float* __restrict__ phi,
             const float* __restrict__ hpre_base, const float* __restrict__ hpost_base,
             const float* __restrict__ hres_base,
             const float* __restrict__ a_pre, const float* __restrict__ a_post,
             const float* __restrict__ a_res,
             float* __restrict__ W, float* __restrict__ hpost_out) {
    __shared__ float xs[ND];       // 16 KB
    __shared__ float red[256];
    __shared__ float dsum[NDELTA];

    const int b = blockIdx.x;
    const int tid = threadIdx.x;

    // finish the mean over S; accumulate sum of squares
    float sq = 0.f;
    for (int c = tid; c < ND; c += 256) {
        float s = 0.f;
        #pragma unroll
        for (int ch = 0; ch < NCHUNK; ++ch)
            s += partial[(long)(b * NCHUNK + ch) * ND + c];
        float xv = s * (1.0f / (float)S_);
        xs[c] = xv;
        sq += xv * xv;
    }
    red[tid] = sq;
    __syncthreads();
    for (int off = 128; off > 0; off >>= 1) {
        if (tid < off) red[tid] += red[tid + off];
        __syncthreads();
    }
    const float rms   = sqrtf(red[0] * (1.0f / (float)ND) + 1e-6f);
    const float scale = 1.0f / rms;
    __syncthreads();

    // delta = (x/rms) @ phi  (phi is (4096, 24) row-major)
    float d[NDELTA];
    #pragma unroll
    for (int k = 0; k < NDELTA; ++k) d[k] = 0.f;
    for (int c = tid; c < ND; c += 256) {
        const float xv = xs[c] * scale;
        const float* ph = phi + (long)c * NDELTA;
        #pragma unroll
        for (int k = 0; k < NDELTA; ++k) d[k] += xv * ph[k];
    }
    for (int k = 0; k < NDELTA; ++k) {
        red[tid] = d[k];
        __syncthreads();
        for (int off = 128; off > 0; off >>= 1) {
            if (tid < off) red[tid] += red[tid + off];
            __syncthreads();
        }
        if (tid == 0) dsum[k] = red[0];
        __syncthreads();
    }

    if (tid == 0) {
        const float ap = a_pre[0], aq = a_post[0], ar = a_res[0];
        // H_pre: sigmoid then normalize
        float hpre[NS], hsum = 0.f;
        for (int i = 0; i < NS; ++i) {
            hpre[i] = 1.0f / (1.0f + expf(-(hpre_base[i] + ap * dsum[i])));
            hsum += hpre[i];
        }
        const float hinv = 1.0f / (hsum + 1e-8f);
        for (int i = 0; i < NS; ++i) hpre[i] *= hinv;
        // H_post: sigmoid
        float hpost[NS];
        for (int i = 0; i < NS; ++i)
            hpost[i] = 1.0f / (1.0f + expf(-(hpost_base[i] + aq * dsum[NS + i])));
        // Sinkhorn on 4x4
        float P[NS * NS];
        for (int i = 0; i < NS; ++i)
            for (int j = 0; j < NS; ++j)
                P[i * NS + j] = expf(hres_base[i * NS + j] + ar * dsum[2 * NS + i * NS + j]);
        for (int it = 0; it < 20; ++it) {
            for (int i = 0; i < NS; ++i) {
                float rs = P[i*NS+0] + P[i*NS+1] + P[i*NS+2] + P[i*NS+3];
                float r = 1.0f / (rs + 1e-6f);
                for (int j = 0; j < NS; ++j) P[i * NS + j] *= r;
            }
            for (int j = 0; j < NS; ++j) {
                float cs = P[0*NS+j] + P[1*NS+j] + P[2*NS+j] + P[3*NS+j];
                float r = 1.0f / (cs + 1e-6f);
                for (int i = 0; i < NS; ++i) P[i * NS + j] *= r;
            }
        }
        // write padded 16x4 A-matrix: row0 = H_pre, rows1..4 = H_res, rest 0
        float* Wb = W + b * 64;
        for (int m = 0; m < 16; ++m)
            for (int k = 0; k < NS; ++k)
                Wb[m * 4 + k] = (m == 0) ? hpre[k] : ((m <= 4) ? P[(m - 1) * NS + k] : 0.f);
        for (int i = 0; i < NS; ++i) hpost_out[b * 4 + i] = hpost[i];
    }
}

// ---------------------------------------------------------------------------
// Phase 3: one wave per (b,s) row; 64 d-tiles each, unrolled x4.
// One V_WMMA_F32_16X16X4_F32 per 16-column tile:
//   A(16x4) = padded [H_pre; H_res] per batch (hoisted; b is wave-constant),
//   B(4x16) = H[b,s,:,d0..d0+15] (lane halves supply K=0,1 / K=2,3),
//   C(16x16) rows 1..4 lanes 0-15 = H_post[i] * branch_output (mask-multiplied
//   so EXEC stays all-ones through the WMMA).
// All addresses = wave base + compile-time immediate offsets.
// ---------------------------------------------------------------------------
__global__ __launch_bounds__(256)
void k_apply(const float* __restrict__ H, const float* __restrict__ bo,
             const float* __restrict__ W, const float* __restrict__ hpostArr,
             float* __restrict__ out) {
    const int lane = threadIdx.x & 31;
    const int col  = lane & 15;
    const int half = lane >> 4;
    const int wv   = (int)((blockIdx.x * blockDim.x + threadIdx.x) >> 5);
    if (wv >= NBS) return;

    const int b = wv >> 12;                 // wv / S_
    // wave-constant operands (b fixed): A matrix + masked H_post row scales
    const v2f Am = *(const v2f*)(W + b * 64 + col * 4 + 2 * half);
    const float4 hp = *(const float4*)(hpostArr + b * 4);
    const float maskf = (half == 0) ? 1.0f : 0.0f;   // rows 8-15 of C must be 0
    const float hp0 = hp.x * maskf, hp1 = hp.y * maskf;
    const float hp2 = hp.z * maskf, hp3 = hp.w * maskf;

    // per-wave base pointers; everything below uses immediate offsets
    const float* hrow = H  + (long)wv * ND + (long)(2 * half) * D_ + col;
    const float* brow = bo + (long)wv * D_ + col;   // both halves: same 64B line
    float*       orow = out + (long)wv * 5 * D_ + col;

    for (int dt = 0; dt < 64; dt += 4) {
        const int d0 = dt << 4;
        v8f Dm[4];
        #pragma unroll
        for (int u = 0; u < 4; ++u) {
            const int dd = d0 + (u << 4);
            v2f Bm;
            Bm.x = hrow[dd];            // K = 2*half
            Bm.y = hrow[dd + D_];       // K = 2*half+1
            const float bov = brow[dd];
            v8f C = {0.f, 0.f, 0.f, 0.f, 0.f, 0.f, 0.f, 0.f};
            C[1] = hp0 * bov;
            C[2] = hp1 * bov;
            C[3] = hp2 * bov;
            C[4] = hp3 * bov;
            Dm[u] = __builtin_amdgcn_wmma_f32_16x16x4_f32(
                false, Am, false, Bm, (short)0, C, false, false);
        }
        if (half == 0) {
            #pragma unroll
            for (int u = 0; u < 4; ++u) {
                const int dd = d0 + (u << 4);
                orow[dd]          = Dm[u][0];   // branch_input
                orow[dd + 1 * D_] = Dm[u][1];   // H_new rows
                orow[dd + 2 * D_] = Dm[u][2];
                orow[dd + 3 * D_] = Dm[u][3];
                orow[dd + 4 * D_] = Dm[u][4];
            }
        }
    }
}

// ---------------------------------------------------------------------------
extern "C" void kernel_launch(void* const* d_in, const int* in_sizes, int n_in,
                              void* d_out, int out_size, void* d_ws, size_t ws_size,
                              hipStream_t stream) {
    const float* H    = (const float*)d_in[0];
    const float* bo   = (const float*)d_in[1];
    const float* phi  = (const float*)d_in[2];
    const float* hprb = (const float*)d_in[3];
    const float* hpob = (const float*)d_in[4];
    const float* hreb = (const float*)d_in[5];
    const float* apre = (const float*)d_in[6];
    const float* apost= (const float*)d_in[7];
    const float* ares = (const float*)d_in[8];
    float* out = (float*)d_out;

    // small gate scratch always in ws (phase 3 reads it while writing d_out)
    float* wsf   = (float*)d_ws;
    float* W     = wsf;            // 256 floats (4 * 16x4)
    float* hpost = wsf + 256;      // 16 floats (16B aligned)

    // 1 MB partial buffer: ws if it fits, else tail of d_out (consumed by
    // phase 2 before phase 3 overwrites d_out)
    const long PART_ELEMS = (long)B_ * NCHUNK * ND;   // 262144
    float* partial;
    if (ws_size >= (size_t)(1024 + PART_ELEMS) * sizeof(float))
        partial = wsf + 1024;
    else
        partial = out + ((long)out_size - PART_ELEMS);

    dim3 g1(4, NCHUNK, B_);
    k_partial<<<g1, 256, 0, stream>>>(H, partial);

    k_gates<<<B_, 256, 0, stream>>>(partial, phi, hprb, hpob, hreb,
                                    apre, apost, ares, W, hpost);

    // one wave per (b,s): 16384 waves = 2048 blocks * 8 waves
    k_apply<<<NBS / 8, 256, 0, stream>>>(H, bo, W, hpost, out);
}